// GATConv_72378788872284
// MI455X (gfx1250) — compile-verified
//
#include <hip/hip_runtime.h>
#include <hip/hip_bf16.h>

#define NN      32768
#define DEG     16
#define NHEADS  4
#define DOUT    64
#define HD      256          // NHEADS*DOUT == IN_FEATS
#define INF_    256
#define SLOPE   0.2f
#define GM      0.1f
#define CM      0.1f

typedef float v2f __attribute__((ext_vector_type(2)));
typedef float v8f __attribute__((ext_vector_type(8)));

__device__ __forceinline__ float leaky(float x) { return x >= 0.0f ? x : SLOPE * x; }
__device__ __forceinline__ float relu0(float x) { return x > 0.0f ? x : 0.0f; }

__device__ __forceinline__ float wave_reduce_add(float v) {
    #pragma unroll
    for (int o = 16; o >= 1; o >>= 1) v += __shfl_xor(v, o, 32);
    return v;
}

// ---------------------------------------------------------------------------
// Kernel 0: zero the two loss accumulators at the tail of d_out
// ---------------------------------------------------------------------------
__global__ void zero_tail_kernel(float* __restrict__ out) {
    if (threadIdx.x < 2) out[(size_t)NN * HD + threadIdx.x] = 0.0f;
}

// ---------------------------------------------------------------------------
// Kernel 1: feat_src = feat @ W.T   via V_WMMA_F32_16X16X4_F32
//   feat: [NN, 256] f32, W: [256, 256] f32 (row j = output channel)
//   out[n][j] = sum_k feat[n][k] * W[j][k]
// Block = 256 threads (8 waves). Wave w -> rows [bx*128 + w*16, +16),
// cols [by*64, +64) as four 16x16 tiles sharing the A fragment.
// ---------------------------------------------------------------------------
__global__ void __launch_bounds__(256)
gemm_wmma_f32_kernel(const float* __restrict__ feat,
                     const float* __restrict__ W,
                     float* __restrict__ out) {
    const int lane = threadIdx.x & 31;
    const int wave = threadIdx.x >> 5;
    const int rowBase = blockIdx.x * 128 + wave * 16;
    const int colBase = blockIdx.y * 64;

    const int mr = lane & 15;               // A row / B col within tile
    const int kk = (lane < 16) ? 0 : 2;     // K sub-offset per ISA 16x4 layout

    const float* aPtr  = feat + (size_t)(rowBase + mr) * INF_ + kk;
    const float* bPtr0 = W + (size_t)(colBase + mr +  0) * INF_ + kk;
    const float* bPtr1 = W + (size_t)(colBase + mr + 16) * INF_ + kk;
    const float* bPtr2 = W + (size_t)(colBase + mr + 32) * INF_ + kk;
    const float* bPtr3 = W + (size_t)(colBase + mr + 48) * INF_ + kk;

    v8f acc0 = {}, acc1 = {}, acc2 = {}, acc3 = {};

    for (int k0 = 0; k0 < INF_; k0 += 4) {
        if ((k0 & 28) == 0)                      // prefetch one line ahead
            __builtin_prefetch(aPtr + k0 + 32, 0, 0);
        v2f a  = *(const v2f*)(aPtr  + k0);
        v2f b0 = *(const v2f*)(bPtr0 + k0);
        v2f b1 = *(const v2f*)(bPtr1 + k0);
        v2f b2 = *(const v2f*)(bPtr2 + k0);
        v2f b3 = *(const v2f*)(bPtr3 + k0);
        acc0 = __builtin_amdgcn_wmma_f32_16x16x4_f32(false, a, false, b0, (short)0, acc0, false, false);
        acc1 = __builtin_amdgcn_wmma_f32_16x16x4_f32(false, a, false, b1, (short)0, acc1, false, false);
        acc2 = __builtin_amdgcn_wmma_f32_16x16x4_f32(false, a, false, b2, (short)0, acc2, false, false);
        acc3 = __builtin_amdgcn_wmma_f32_16x16x4_f32(false, a, false, b3, (short)0, acc3, false, false);
    }

    // C/D layout: VGPR v -> row v (lanes 0-15) / row v+8 (lanes 16-31), col = lane&15
    const int rOff = (lane < 16) ? 0 : 8;
    const int c    = lane & 15;
    #pragma unroll
    for (int v = 0; v < 8; ++v) {
        float* o = out + (size_t)(rowBase + v + rOff) * HD + colBase + c;
        o[0]  = acc0[v];
        o[16] = acc1[v];
        o[32] = acc2[v];
        o[48] = acc3[v];
    }
}

// ---------------------------------------------------------------------------
// Kernel 2: el[n,h] = <feat_src[n,h,:], attn_l[h,:]>, er likewise
// One thread per (n,h).
// ---------------------------------------------------------------------------
__global__ void __launch_bounds__(256)
attn_proj_kernel(const float* __restrict__ fs,
                 const float* __restrict__ attn_l,
                 const float* __restrict__ attn_r,
                 float* __restrict__ el, float* __restrict__ er) {
    const int t = blockIdx.x * blockDim.x + threadIdx.x;   // n*4 + h
    if (t >= NN * NHEADS) return;
    const int h = t & 3;
    const int n = t >> 2;
    const float4* fv = (const float4*)(fs + (size_t)n * HD + h * DOUT);
    const float4* lv = (const float4*)(attn_l + h * DOUT);
    const float4* rv = (const float4*)(attn_r + h * DOUT);
    float sl = 0.0f, sr = 0.0f;
    #pragma unroll
    for (int i = 0; i < DOUT / 4; ++i) {
        float4 f = fv[i], l = lv[i], r = rv[i];
        sl += f.x * l.x + f.y * l.y + f.z * l.z + f.w * l.w;
        sr += f.x * r.x + f.y * r.y + f.z * r.z + f.w * r.w;
    }
    el[t] = sl;
    er[t] = sr;
}

// ---------------------------------------------------------------------------
// Kernel 3: per-dst-node phase. One wave32 per node, 8 nodes per block.
//   pw/nw edge scores -> margin losses (atomic), edge softmax, weighted agg.
//   dst[k] == k/DEG by construction; neg_dst[k] == neg_perm[k]/DEG.
// ---------------------------------------------------------------------------
__global__ void __launch_bounds__(256)
gat_node_kernel(const float* __restrict__ fs,
                const float* __restrict__ el,
                const float* __restrict__ er,
                const int* __restrict__ src,
                const int* __restrict__ neg_perm,
                const int* __restrict__ label,
                float* __restrict__ out) {
    __shared__ int   sidx[8][DEG];
    __shared__ int   sadj[8][DEG];
    __shared__ float spw[8][DEG][NHEADS];
    __shared__ float snw[8][DEG][NHEADS];
    __shared__ float sat[8][DEG][NHEADS];

    const int wave = threadIdx.x >> 5;
    const int lane = threadIdx.x & 31;
    const int n    = blockIdx.x * 8 + wave;

    // ---- phase 1: edge scores ----
    if (lane < DEG) {
        const int k  = n * DEG + lane;
        const int s  = src[k];
        const int nd = neg_perm[k] >> 4;            // neg dst node
        sidx[wave][lane] = s;
        sadj[wave][lane] = (label[s] == label[n]) ? 1 : 0;
        #pragma unroll
        for (int h = 0; h < NHEADS; ++h) {
            const float elv = el[s * NHEADS + h];
            spw[wave][lane][h] = leaky(elv + er[n  * NHEADS + h]);
            snw[wave][lane][h] = leaky(elv + er[nd * NHEADS + h]);
        }
    }
    __syncthreads();

    // ---- phase 2: margin losses over (i,j,h) = 16*16*4 = 1024 pairs ----
    float lg = 0.0f, lb = 0.0f;
    for (int t = lane; t < DEG * DEG * NHEADS; t += 32) {
        const int h = t & 3;
        const int j = (t >> 2) & 15;     // pos index (pw) / 'b'
        const int i = t >> 6;            // neg index (nw) / 'a'
        lg += relu0(snw[wave][i][h] + GM - spw[wave][j][h]);
        if (sadj[wave][i] && !sadj[wave][j])
            lb += relu0(spw[wave][j][h] + CM - spw[wave][i][h]);
    }
    lg = wave_reduce_add(lg);
    lb = wave_reduce_add(lb);
    if (lane == 0) {
        const float inv = 1.0f / ((float)NN * (float)NHEADS);
        atomicAdd(out + (size_t)NN * HD,     lg * inv);
        atomicAdd(out + (size_t)NN * HD + 1, lb * inv);
    }

    // ---- phase 3: edge softmax over DEG, one lane per head ----
    if (lane < NHEADS) {
        const int h = lane;
        float m = -3.4e38f;
        #pragma unroll
        for (int d = 0; d < DEG; ++d) m = fmaxf(m, spw[wave][d][h]);
        float s = 0.0f;
        #pragma unroll
        for (int d = 0; d < DEG; ++d) {
            const float e = __expf(spw[wave][d][h] - m);
            sat[wave][d][h] = e;
            s += e;
        }
        const float invs = 1.0f / s;
        #pragma unroll
        for (int d = 0; d < DEG; ++d) sat[wave][d][h] *= invs;
    }
    __syncthreads();

    // ---- phase 4: attention-weighted aggregation ----
    // lane -> head = lane/8, feature block = (lane%8)*8 (8 floats = 2 float4)
    const int h  = lane >> 3;
    const int fb = (lane & 7) * 8;
    float4 acc0 = {0, 0, 0, 0}, acc1 = {0, 0, 0, 0};
    #pragma unroll
    for (int d = 0; d < DEG; ++d) {
        const int s   = sidx[wave][d];
        const float w = sat[wave][d][h];
        const float4* p = (const float4*)(fs + (size_t)s * HD + h * DOUT + fb);
        const float4 x0 = p[0], x1 = p[1];
        acc0.x += w * x0.x; acc0.y += w * x0.y; acc0.z += w * x0.z; acc0.w += w * x0.w;
        acc1.x += w * x1.x; acc1.y += w * x1.y; acc1.z += w * x1.z; acc1.w += w * x1.w;
    }
    float4* o = (float4*)(out + (size_t)n * HD + h * DOUT + fb);
    o[0] = acc0;
    o[1] = acc1;
}

// ---------------------------------------------------------------------------
extern "C" void kernel_launch(void* const* d_in, const int* in_sizes, int n_in,
                              void* d_out, int out_size, void* d_ws, size_t ws_size,
                              hipStream_t stream) {
    (void)in_sizes; (void)n_in; (void)out_size; (void)ws_size;
    const float* feat     = (const float*)d_in[0];
    const int*   label    = (const int*)  d_in[1];
    const int*   src      = (const int*)  d_in[2];
    /* d_in[3] = dst: implicit (k/DEG), unused */
    const int*   neg_perm = (const int*)  d_in[4];
    const float* W        = (const float*)d_in[5];
    const float* attn_l   = (const float*)d_in[6];
    const float* attn_r   = (const float*)d_in[7];
    float*       out      = (float*)d_out;

    float* fs = (float*)d_ws;                       // [NN, 256]  feat_src
    float* el = fs + (size_t)NN * HD;               // [NN, 4]
    float* er = el + (size_t)NN * NHEADS;           // [NN, 4]

    zero_tail_kernel<<<1, 64, 0, stream>>>(out);
    gemm_wmma_f32_kernel<<<dim3(NN / 128, HD / 64), 256, 0, stream>>>(feat, W, fs);
    attn_proj_kernel<<<(NN * NHEADS) / 256, 256, 0, stream>>>(fs, attn_l, attn_r, el, er);
    gat_node_kernel<<<NN / 8, 256, 0, stream>>>(fs, el, er, src, neg_perm, label, out);
}